// AdditiveAttn_36266703847882
// MI455X (gfx1250) — compile-verified
//
#include <hip/hip_runtime.h>
#include <math.h>

// Problem constants (match reference)
#define NN 50000
#define EE 500000
#define HH 8
#define DD 16
// H*D = 128, 2*H*D = 256

typedef __attribute__((ext_vector_type(2))) float v2f;
typedef __attribute__((ext_vector_type(8))) float v8f;

__device__ __forceinline__ v8f wmma_k4(v2f a, v2f b, v8f c) {
  // V_WMMA_F32_16X16X4_F32 : D = A(16x4) * B(4x16) + C(16x16), fp32
  return __builtin_amdgcn_wmma_f32_16x16x4_f32(
      /*neg_a=*/false, a, /*neg_b=*/false, b,
      /*c_mod=*/(short)0, c, /*reuse_a=*/false, /*reuse_b=*/false);
}

// --- async global->LDS copy (CDNA5 GLOBAL_LOAD_ASYNC_TO_LDS_B128) -----------
#if defined(__has_builtin)
#if __has_builtin(__builtin_amdgcn_global_load_async_to_lds_b128) && \
    __has_builtin(__builtin_amdgcn_s_wait_asynccnt)
#define HAVE_ASYNC_LDS 1
#endif
#endif

#ifdef HAVE_ASYNC_LDS
// Builtin signature (from compiler diagnostic): param0 = int4 AS(1)*,
// param1 = int4 AS(3)*, then two constant ints (offset, cpol).
typedef __attribute__((__vector_size__(4 * sizeof(int)))) int v4i_;
typedef __attribute__((address_space(1))) v4i_* gptr_v4i;
typedef __attribute__((address_space(3))) v4i_* lptr_v4i;
#endif

// Stage a 16x128 fp32 tile (row-major, LDS stride 132 floats to avoid bank
// conflicts) into LDS using 128-bit async copies when available.
// Call with all 256 threads; __syncthreads() performed inside.
__device__ __forceinline__ void stage_tile_16x128(const float* __restrict__ src,
                                                  size_t rowBase,
                                                  float* lds /*16*132*/) {
#ifdef HAVE_ASYNC_LDS
  // 512 float4 chunks; 2 per thread. Rows are contiguous (pad only between rows).
#pragma unroll
  for (int j = 0; j < 2; ++j) {
    const int chunk = threadIdx.x + j * 256;
    const int r = chunk >> 5;
    const int c4 = (chunk & 31) * 4;
    const float* g = &src[(rowBase + r) * 128 + c4];
    __builtin_amdgcn_global_load_async_to_lds_b128(
        (gptr_v4i)g, (lptr_v4i)&lds[r * 132 + c4], 0, 0);
  }
  __builtin_amdgcn_s_wait_asynccnt(0);
#else
#pragma unroll
  for (int j = 0; j < 2; ++j) {
    const int chunk = threadIdx.x + j * 256;
    const int r = chunk >> 5;
    const int c4 = (chunk & 31) * 4;
    const float4 v = *(const float4*)&src[(rowBase + r) * 128 + c4];
    *(float4*)&lds[r * 132 + c4] = v;
  }
#endif
  __syncthreads();
}

// ---------------------------------------------------------------------------
// Kernel 0: zero workspace accumulators
// ---------------------------------------------------------------------------
__global__ void zero_ws_kernel(float* p, long n) {
  long t = (long)blockIdx.x * blockDim.x + threadIdx.x;
  if (t < n) p[t] = 0.0f;
}

// ---------------------------------------------------------------------------
// Kernel 1: Qh/Kh/Vh = x @ W + b   (blockIdx.y selects which of Q/K/V)
// block = 256 threads = 8 waves; wave w -> output col tile w (16 cols)
// blockIdx.x -> 16-node row tile. x tile staged in LDS.
// ---------------------------------------------------------------------------
__global__ void node_qkv_kernel(const float* __restrict__ x,
                                const float* __restrict__ WQ, const float* __restrict__ bQ,
                                const float* __restrict__ WK, const float* __restrict__ bK,
                                const float* __restrict__ WV, const float* __restrict__ bV,
                                float* __restrict__ Qh, float* __restrict__ Kh,
                                float* __restrict__ Vh) {
  __shared__ float xt[16 * 132];
  const int mBase = blockIdx.x * 16;
  const int which = blockIdx.y;
  const float* W    = (which == 0) ? WQ : (which == 1) ? WK : WV;
  const float* bias = (which == 0) ? bQ : (which == 1) ? bK : bV;
  float* out        = (which == 0) ? Qh : (which == 1) ? Kh : Vh;

  stage_tile_16x128(x, (size_t)mBase, xt);

  const int wave = threadIdx.x >> 5;
  const int lane = threadIdx.x & 31;
  const int hi = lane >> 4;   // lane half selects K sub-pair
  const int n  = lane & 15;   // output column within tile / A-row m
  const int colBase = wave * 16;

  v8f acc = {};
#pragma unroll
  for (int k = 0; k < 128; k += 4) {
    v2f a, b;
    a.x = xt[n * 132 + k + 2 * hi];
    a.y = xt[n * 132 + k + 2 * hi + 1];
    const int r0 = (k + 2 * hi) * 128 + colBase + n;
    b.x = W[r0];
    b.y = W[r0 + 128];
    acc = wmma_k4(a, b, acc);
  }
  const float bb = bias[colBase + n];
#pragma unroll
  for (int v = 0; v < 8; ++v) {
    out[(size_t)(mBase + v + 8 * hi) * 128 + colBase + n] = acc[v] + bb;
  }
}

// ---------------------------------------------------------------------------
// Kernel 2 (fused): per 16-edge tile
//   Ex = edge_attr @ WE + bE  (WMMA, never materialized)
//   score2 = signed-sqrt(Ex[:, :D] * Ex[:, D:])
//   conn = Kh[src] + Qh[dst] + score2          -> written to d_out
//   score = clamp(conn . Aw, +-5); ex = exp(score)   (max-free softmax: safe
//   because score is clamped to [-5,5])
//   denom[dst,h] += ex (atomic); exsc[e,h] = ex
// block = 256 threads = 8 waves; wave w = head w. blockIdx.x = edge tile.
// ---------------------------------------------------------------------------
__global__ void edge_fused_kernel(const float* __restrict__ ea,
                                  const int* __restrict__ eidx,
                                  const float* __restrict__ WE,
                                  const float* __restrict__ bE,
                                  const float* __restrict__ Qh,
                                  const float* __restrict__ Kh,
                                  const float* __restrict__ Aw,
                                  float* __restrict__ connOut,
                                  float* __restrict__ exsc,
                                  float* __restrict__ denom) {
  __shared__ float et[16 * 132];
  const int eBase = blockIdx.x * 16;

  stage_tile_16x128(ea, (size_t)eBase, et);

  const int h    = threadIdx.x >> 5;  // head handled by this wave
  const int lane = threadIdx.x & 31;
  const int hi = lane >> 4;
  const int n  = lane & 15;           // feature index d within head

  v8f acc1 = {}, acc2 = {};
#pragma unroll
  for (int k = 0; k < 128; k += 4) {
    v2f a, b1, b2;
    a.x = et[n * 132 + k + 2 * hi];
    a.y = et[n * 132 + k + 2 * hi + 1];
    const int r0 = (k + 2 * hi) * 256 + h * 32 + n;
    b1.x = WE[r0];        b1.y = WE[r0 + 256];
    b2.x = WE[r0 + 16];   b2.y = WE[r0 + 256 + 16];
    acc1 = wmma_k4(a, b1, acc1);
    acc2 = wmma_k4(a, b2, acc2);
  }

  const float be1 = bE[h * 32 + n];
  const float be2 = bE[h * 32 + 16 + n];
  const float aw  = Aw[n * 8 + h];   // Aw shape (D,H,1)

#pragma unroll
  for (int v = 0; v < 8; ++v) {
    const int e  = eBase + v + 8 * hi;
    const int s  = eidx[e];
    const int dd = eidx[EE + e];
    const float p1 = acc1[v] + be1;
    const float p2 = acc2[v] + be2;
    const float s2 = p1 * p2;
    const float sc2 = (s2 >= 0.0f) ? sqrtf(s2) : -sqrtf(-s2);
    const float conn = Kh[(size_t)s * 128 + h * 16 + n] +
                       Qh[(size_t)dd * 128 + h * 16 + n] + sc2;
    connOut[(size_t)e * 128 + h * 16 + n] = conn;

    // reduce conn*Aw over the 16 feature lanes of this half-wave
    float part = conn * aw;
    part += __shfl_xor(part, 1, 32);
    part += __shfl_xor(part, 2, 32);
    part += __shfl_xor(part, 4, 32);
    part += __shfl_xor(part, 8, 32);
    float score = fminf(5.0f, fmaxf(-5.0f, part));
    float ex = expf(score);
    if (n == v) {             // exactly one lane per (edge, head)
      exsc[(size_t)e * 8 + h] = ex;
      atomicAdd(&denom[(size_t)dd * 8 + h], ex);
    }
  }
}

// ---------------------------------------------------------------------------
// Kernel 3: scatter pass.  One thread per (edge, head).
//   attn = ex / (denom[dst,h] + 1e-16)
//   VoAcc[dst,h,:] += Vh[src,h,:] * attn ;  rowV[dst,h,:] += conn[e,h,:] * attn
// ---------------------------------------------------------------------------
__global__ void edge_scatter_kernel(const int* __restrict__ eidx,
                                    const float* __restrict__ Vh,
                                    const float* __restrict__ connOut,
                                    const float* __restrict__ exsc,
                                    const float* __restrict__ denom,
                                    float* __restrict__ VoAcc,
                                    float* __restrict__ rowV) {
  const int t = blockIdx.x * 256 + threadIdx.x;  // t < E*H
  const int e = t >> 3;
  const int h = t & 7;
  const int s  = eidx[e];
  const int dd = eidx[EE + e];
  const float attn = exsc[t] / (denom[(size_t)dd * 8 + h] + 1e-16f);
  const float* vrow = &Vh[(size_t)s * 128 + h * 16];
  const float* crow = &connOut[(size_t)e * 128 + h * 16];
  float* vo = &VoAcc[(size_t)dd * 128 + h * 16];
  float* rv = &rowV[(size_t)dd * 128 + h * 16];
#pragma unroll
  for (int d = 0; d < 16; ++d) {
    atomicAdd(&vo[d], vrow[d] * attn);
    atomicAdd(&rv[d], crow[d] * attn);
  }
}

// ---------------------------------------------------------------------------
// Kernel 4: Vo = Qh + VoAcc + rowV @ VeRow (per-head 16x16 GEMM via WMMA)
// block = 256 = 8 waves; wave w = head w. blockIdx.x = 16-node tile.
// ---------------------------------------------------------------------------
__global__ void node_final_kernel(const float* __restrict__ Qh,
                                  const float* __restrict__ VoAcc,
                                  const float* __restrict__ rowV,
                                  const float* __restrict__ VeRow,
                                  float* __restrict__ VoOut) {
  const int mBase = blockIdx.x * 16;
  const int h    = threadIdx.x >> 5;
  const int lane = threadIdx.x & 31;
  const int hi = lane >> 4;
  const int n  = lane & 15;

  v8f acc = {};
#pragma unroll
  for (int kk = 0; kk < 16; kk += 4) {
    v2f a, b;
    const int kA = kk + 2 * hi;
    const size_t abase = (size_t)(mBase + n) * 128 + h * 16 + kA;
    a.x = rowV[abase];
    a.y = rowV[abase + 1];
    b.x = VeRow[(kA * 8 + h) * 16 + n];         // VeRow shape (D,H,D)
    b.y = VeRow[((kA + 1) * 8 + h) * 16 + n];
    acc = wmma_k4(a, b, acc);
  }
#pragma unroll
  for (int v = 0; v < 8; ++v) {
    const size_t idx = (size_t)(mBase + v + 8 * hi) * 128 + h * 16 + n;
    VoOut[idx] = Qh[idx] + VoAcc[idx] + acc[v];
  }
}

// ---------------------------------------------------------------------------
// Launcher
// ---------------------------------------------------------------------------
extern "C" void kernel_launch(void* const* d_in, const int* in_sizes, int n_in,
                              void* d_out, int out_size, void* d_ws, size_t ws_size,
                              hipStream_t stream) {
  const float* x     = (const float*)d_in[0];
  const float* ea    = (const float*)d_in[1];
  const int*   eidx  = (const int*)  d_in[2];
  const float* WQ    = (const float*)d_in[3];
  const float* bQ    = (const float*)d_in[4];
  const float* WK    = (const float*)d_in[5];
  const float* bK    = (const float*)d_in[6];
  const float* WE    = (const float*)d_in[7];
  const float* bE    = (const float*)d_in[8];
  const float* WV    = (const float*)d_in[9];
  const float* bV    = (const float*)d_in[10];
  const float* Aw    = (const float*)d_in[11];
  const float* VeRow = (const float*)d_in[12];

  float* out     = (float*)d_out;
  float* VoOut   = out;                       // [N, 128]
  float* connOut = out + (size_t)NN * 128;    // [E, 128]

  // workspace layout (floats): Qh | Kh | Vh | denom | exsc | VoAcc | rowV
  float* ws    = (float*)d_ws;
  float* Qh    = ws;
  float* Kh    = Qh    + (size_t)NN * 128;
  float* Vh    = Kh    + (size_t)NN * 128;
  float* denom = Vh    + (size_t)NN * 128;
  float* exsc  = denom + (size_t)NN * 8;
  float* VoAcc = exsc  + (size_t)EE * 8;
  float* rowV  = VoAcc + (size_t)NN * 128;

  // zero denom + exsc + VoAcc + rowV (contiguous)
  const long nzero = (long)NN * 8 + (long)EE * 8 + 2L * NN * 128;  // 17.2M
  zero_ws_kernel<<<(int)((nzero + 255) / 256), 256, 0, stream>>>(denom, nzero);

  // 1) Qh/Kh/Vh GEMMs (N = 3125*16 exactly)
  node_qkv_kernel<<<dim3(NN / 16, 3), 256, 0, stream>>>(
      x, WQ, bQ, WK, bK, WV, bV, Qh, Kh, Vh);

  // 2) fused edge GEMM + score + softmax numerator (E = 31250*16 exactly)
  edge_fused_kernel<<<EE / 16, 256, 0, stream>>>(
      ea, eidx, WE, bE, Qh, Kh, Aw, connOut, exsc, denom);

  // 3) weighted scatter-add (E*H / 256 = 15625 exactly)
  edge_scatter_kernel<<<(EE * HH) / 256, 256, 0, stream>>>(
      eidx, Vh, connOut, exsc, denom, VoAcc, rowV);

  // 4) final output projection + residual
  node_final_kernel<<<NN / 16, 256, 0, stream>>>(
      Qh, VoAcc, rowV, VeRow, VoOut);
}